// GINWithVirtualNode_36335423324485
// MI455X (gfx1250) — compile-verified
//
#include <hip/hip_runtime.h>
#include <hip/hip_bf16.h>

typedef __attribute__((ext_vector_type(2))) float v2f;
typedef __attribute__((ext_vector_type(8))) float v8f;

#define BN_EPS_F 1e-5f

// ---------------------------------------------------------------------------
// WMMA f32 GEMM:  Y[M,Hout] = A[M,K] @ W[K,Hout] + bias
// One wave computes a 32x32 block = 2x2 grid of 16x16 V_WMMA_F32_16X16X4_F32
// tiles (exact f32).  A/B fragments are each reused across 2 WMMAs -> 1.5
// VMEM loads per WMMA instead of 3.  Workgroup = 256 threads = 8 waves
// arranged 2 (M) x 4 (N) -> 64 x 128 output block.
// Optional fused epilogue: column sum / sum-of-squares (for BatchNorm) via
// LDS ds_add_f32 reduction + one global atomic per column per workgroup.
// ---------------------------------------------------------------------------
template <bool STATS, bool RELU>
__global__ __launch_bounds__(256) void gemm_wmma_f32(
    const float* __restrict__ A, const float* __restrict__ W,
    const float* __restrict__ bias, float* __restrict__ Y,
    int M, int K, int Hout,
    float* __restrict__ colsum, float* __restrict__ colsq) {
  __shared__ float ssum[128];
  __shared__ float ssq[128];
  if (STATS) {
    if (threadIdx.x < 128) { ssum[threadIdx.x] = 0.f; ssq[threadIdx.x] = 0.f; }
    __syncthreads();
  }
  const int tilesN = Hout >> 7;                 // 128 cols per workgroup
  const int blockM = blockIdx.x / tilesN;
  const int blockN = blockIdx.x % tilesN;
  const int wave   = threadIdx.x >> 5;
  const int lane   = threadIdx.x & 31;
  const int wm     = wave & 1;                  // 2 waves in M
  const int wn     = wave >> 1;                 // 4 waves in N
  const int lm     = lane & 15;
  const int lhalf  = lane >> 4;

  const int m0 = blockM * 64 + wm * 32;         // two 16-row subtiles
  const int m1 = m0 + 16;
  const int n0 = blockN * 128 + wn * 32;        // two 16-col subtiles
  const int n1 = n0 + 16;

  int r0 = m0 + lm;
  if (r0 >= M) r0 = M - 1;                      // clamp (stores are guarded)
  int r1 = m1 + lm;
  if (r1 >= M) r1 = M - 1;
  const float* A0 = A + (size_t)r0 * K + 2 * lhalf;
  const float* A1 = A + (size_t)r1 * K + 2 * lhalf;
  const float* W0 = W + (size_t)(2 * lhalf) * Hout + n0 + lm;
  const float* W1 = W0 + 16;

  v8f c00 = {}, c01 = {}, c10 = {}, c11 = {};
  for (int k0 = 0; k0 < K; k0 += 4) {
    // A 16x4 layout: lane m=lane%16; VGPR0 = K {k0, k0+2}, VGPR1 = {k0+1, k0+3}
    v2f a0 = *(const v2f*)(A0 + k0);
    v2f a1 = *(const v2f*)(A1 + k0);
    // B 4x16 layout: VGPR0 = rows {k0, k0+2}, VGPR1 = rows {k0+1, k0+3}
    const size_t w_off = (size_t)k0 * Hout;
    v2f b0, b1;
    b0.x = W0[w_off];
    b0.y = W0[w_off + Hout];
    b1.x = W1[w_off];
    b1.y = W1[w_off + Hout];
    // 4 independent accumulators: no WMMA->WMMA RAW hazard
    c00 = __builtin_amdgcn_wmma_f32_16x16x4_f32(false, a0, false, b0, (short)0,
                                                c00, false, false);
    c01 = __builtin_amdgcn_wmma_f32_16x16x4_f32(false, a0, false, b1, (short)0,
                                                c01, false, false);
    c10 = __builtin_amdgcn_wmma_f32_16x16x4_f32(false, a1, false, b0, (short)0,
                                                c10, false, false);
    c11 = __builtin_amdgcn_wmma_f32_16x16x4_f32(false, a1, false, b1, (short)0,
                                                c11, false, false);
  }

  const int col0 = n0 + lm;
  const int col1 = n1 + lm;
  const float bc0 = bias ? bias[col0] : 0.f;
  const float bc1 = bias ? bias[col1] : 0.f;
  float ls0 = 0.f, lq0 = 0.f, ls1 = 0.f, lq1 = 0.f;
#pragma unroll
  for (int v = 0; v < 8; ++v) {
    // C/D layout: M = subtile_base + v + 8*(lane>=16)
    int ra = m0 + v + 8 * lhalf;
    int rb = m1 + v + 8 * lhalf;
    float v00 = c00[v] + bc0;
    float v01 = c01[v] + bc1;
    float v10 = c10[v] + bc0;
    float v11 = c11[v] + bc1;
    if (RELU) {
      v00 = fmaxf(v00, 0.f); v01 = fmaxf(v01, 0.f);
      v10 = fmaxf(v10, 0.f); v11 = fmaxf(v11, 0.f);
    }
    if (ra < M) {
      Y[(size_t)ra * Hout + col0] = v00;
      Y[(size_t)ra * Hout + col1] = v01;
      if (STATS) { ls0 += v00; lq0 += v00 * v00; ls1 += v01; lq1 += v01 * v01; }
    }
    if (rb < M) {
      Y[(size_t)rb * Hout + col0] = v10;
      Y[(size_t)rb * Hout + col1] = v11;
      if (STATS) { ls0 += v10; lq0 += v10 * v10; ls1 += v11; lq1 += v11 * v11; }
    }
  }
  if (STATS) {
    const int cl0 = wn * 32 + lm;               // columns within WG tile
    const int cl1 = cl0 + 16;
    atomicAdd(&ssum[cl0], ls0);                 // ds_add_f32
    atomicAdd(&ssq[cl0], lq0);
    atomicAdd(&ssum[cl1], ls1);
    atomicAdd(&ssq[cl1], lq1);
    __syncthreads();
    if (threadIdx.x < 128) {
      atomicAdd(&colsum[blockN * 128 + threadIdx.x], ssum[threadIdx.x]);
      atomicAdd(&colsq[blockN * 128 + threadIdx.x], ssq[threadIdx.x]);
    }
  }
}

// ---------------------------------------------------------------------------
// Helper kernels
// ---------------------------------------------------------------------------
__global__ void gather_emb_kernel(const int* __restrict__ atom,
                                  const float* __restrict__ table,
                                  float* __restrict__ out, long total, int D) {
  long idx = (long)blockIdx.x * blockDim.x + threadIdx.x;
  if (idx >= total) return;
  long n = idx / D;
  int k = (int)(idx % D);
  out[idx] = table[(size_t)atom[n] * D + k];
}

__global__ void counts_kernel(const int* __restrict__ batch,
                              float* __restrict__ counts, int N) {
  int n = blockIdx.x * blockDim.x + threadIdx.x;
  if (n < N) atomicAdd(&counts[batch[n]], 1.0f);
}

__global__ void vninit_kernel(float* __restrict__ vn,
                              const float* __restrict__ init, long total, int H) {
  long idx = (long)blockIdx.x * blockDim.x + threadIdx.x;
  if (idx >= total) return;
  vn[idx] = init[idx % H];
}

// agg[dst] += x[src] for each edge; 8 columns per lane (coalesced row slice)
__global__ void edge_agg_kernel(const int* __restrict__ row,
                                const int* __restrict__ col,
                                const float* __restrict__ x,
                                float* __restrict__ agg, int E, int H) {
  long t = (long)blockIdx.x * blockDim.x + threadIdx.x;
  int e = (int)(t >> 5);
  if (e >= E) return;
  int cpt = H >> 5;
  int cc = ((int)t & 31) * cpt;
  int src = col[e], dst = row[e];
  const float* xs = x + (size_t)src * H + cc;
  float* ad = agg + (size_t)dst * H + cc;
  for (int i = 0; i < cpt; ++i) atomicAdd(&ad[i], xs[i]);
}

// h_pre = (1+eps[layer])*x + agg   (written in place over agg)
__global__ void hpre_kernel(const float* __restrict__ x, float* __restrict__ agg,
                            const float* __restrict__ eps, int layer, long total) {
  long idx = (long)blockIdx.x * blockDim.x + threadIdx.x;
  if (idx >= total) return;
  agg[idx] = (1.f + eps[layer]) * x[idx] + agg[idx];
}

// out = relu((y - mean)*rsqrt(var+eps)*g + bt), biased batch stats from sums
__global__ void bn_relu_kernel(const float* __restrict__ Yin,
                               float* __restrict__ out,
                               const float* __restrict__ cs,
                               const float* __restrict__ cq,
                               const float* __restrict__ g,
                               const float* __restrict__ bt, long total, int H,
                               float invM) {
  long idx = (long)blockIdx.x * blockDim.x + threadIdx.x;
  if (idx >= total) return;
  int c = (int)(idx % H);
  float m = cs[c] * invM;
  float var = cq[c] * invM - m * m;
  float rs = rsqrtf(var + BN_EPS_F);
  float val = (Yin[idx] - m) * rs * g[c] + bt[c];
  out[idx] = fmaxf(val, 0.f);
}

// vn += relu(bn(y))
__global__ void vn_acc_kernel(float* __restrict__ vn, const float* __restrict__ y,
                              const float* __restrict__ cs,
                              const float* __restrict__ cq,
                              const float* __restrict__ g,
                              const float* __restrict__ bt, long total, int H,
                              float invM) {
  long idx = (long)blockIdx.x * blockDim.x + threadIdx.x;
  if (idx >= total) return;
  int c = (int)(idx % H);
  float m = cs[c] * invM;
  float var = cq[c] * invM - m * m;
  float rs = rsqrtf(var + BN_EPS_F);
  float val = (y[idx] - m) * rs * g[c] + bt[c];
  vn[idx] += fmaxf(val, 0.f);
}

__global__ void seg_sum_kernel(const int* __restrict__ batch,
                               const float* __restrict__ x,
                               float* __restrict__ s, int N, int H) {
  long t = (long)blockIdx.x * blockDim.x + threadIdx.x;
  int n = (int)(t >> 5);
  if (n >= N) return;
  int cpt = H >> 5;
  int cc = ((int)t & 31) * cpt;
  int b = batch[n];
  const float* xp = x + (size_t)n * H + cc;
  float* sp = s + (size_t)b * H + cc;
  for (int i = 0; i < cpt; ++i) atomicAdd(&sp[i], xp[i]);
}

// mean = counts>0 ? s/counts : 0
__global__ void mean_kernel(const float* __restrict__ s,
                            const float* __restrict__ counts,
                            float* __restrict__ out, long total, int H) {
  long idx = (long)blockIdx.x * blockDim.x + threadIdx.x;
  if (idx >= total) return;
  float cnt = counts[idx / H];
  out[idx] = cnt > 0.f ? s[idx] / cnt : 0.f;
}

// count-weighted column stats of z[B,H]: exact BN stats of z[batch] over N rows
__global__ void wstats_kernel(const float* __restrict__ z,
                              const float* __restrict__ counts,
                              float* __restrict__ wsum, float* __restrict__ wsq,
                              int B, int H) {
  int c = threadIdx.x;                          // blockDim.x == H
  int chunk = (B + gridDim.x - 1) / gridDim.x;
  int r0 = blockIdx.x * chunk;
  int r1 = r0 + chunk;
  if (r1 > B) r1 = B;
  float s = 0.f, q = 0.f;
  for (int r = r0; r < r1; ++r) {
    float w = counts[r];
    float v = z[(size_t)r * H + c];
    s += w * v;
    q += w * v * v;
  }
  atomicAdd(&wsum[c], s);
  atomicAdd(&wsq[c], q);
}

// x[n] += relu(bn(z[batch[n]]))  with weighted stats computed at B granularity
__global__ void v2n_apply_kernel(float* __restrict__ x, const float* __restrict__ z,
                                 const int* __restrict__ batch,
                                 const float* __restrict__ wsum,
                                 const float* __restrict__ wsq,
                                 const float* __restrict__ g,
                                 const float* __restrict__ bt, int N, int H,
                                 float invN) {
  long t = (long)blockIdx.x * blockDim.x + threadIdx.x;
  int n = (int)(t >> 5);
  if (n >= N) return;
  int cpt = H >> 5;
  int cc = ((int)t & 31) * cpt;
  int b = batch[n];
  const float* zp = z + (size_t)b * H;
  float* xp = x + (size_t)n * H;
  for (int i = 0; i < cpt; ++i) {
    int c = cc + i;
    float m = wsum[c] * invN;
    float var = wsq[c] * invN - m * m;
    float rs = rsqrtf(var + BN_EPS_F);
    float val = (zp[c] - m) * rs * g[c] + bt[c];
    xp[c] += fmaxf(val, 0.f);
  }
}

// out[g] = hidden[g,:] . W2 + b2   (Hout=1, trivial dot)
__global__ void pred2_kernel(const float* __restrict__ hidden,
                             const float* __restrict__ W2,
                             const float* __restrict__ b2,
                             float* __restrict__ out, int B, int Hh) {
  int gidx = blockIdx.x * blockDim.x + threadIdx.x;
  if (gidx >= B) return;
  float s = b2[0];
  const float* hp = hidden + (size_t)gidx * Hh;
  for (int k = 0; k < Hh; ++k) s += hp[k] * W2[k];
  out[gidx] = s;
}

// ---------------------------------------------------------------------------
static inline int ceil_div_l(long a, long b) { return (int)((a + b - 1) / b); }

extern "C" void kernel_launch(void* const* d_in, const int* in_sizes, int n_in,
                              void* d_out, int out_size, void* d_ws,
                              size_t ws_size, hipStream_t stream) {
  const int* atom_x     = (const int*)d_in[0];
  const int* edge_index = (const int*)d_in[1];
  const int* batch      = (const int*)d_in[2];
  const float* emb_table = (const float*)d_in[4];
  const float* proj_W    = (const float*)d_in[5];
  const float* proj_b    = (const float*)d_in[6];
  const float* gin_eps   = (const float*)d_in[7];
  const float* gin_W1    = (const float*)d_in[8];
  const float* gin_b1    = (const float*)d_in[9];
  const float* gin_g1    = (const float*)d_in[10];
  const float* gin_bt1   = (const float*)d_in[11];
  const float* gin_W2    = (const float*)d_in[12];
  const float* gin_b2    = (const float*)d_in[13];
  const float* gin_g2    = (const float*)d_in[14];
  const float* gin_bt2   = (const float*)d_in[15];
  const float* vn_init   = (const float*)d_in[16];
  const float* n2v_W     = (const float*)d_in[17];
  const float* n2v_b     = (const float*)d_in[18];
  const float* n2v_g     = (const float*)d_in[19];
  const float* n2v_bt    = (const float*)d_in[20];
  const float* v2n_W     = (const float*)d_in[21];
  const float* v2n_b     = (const float*)d_in[22];
  const float* v2n_g     = (const float*)d_in[23];
  const float* v2n_bt    = (const float*)d_in[24];
  const float* pred_W1   = (const float*)d_in[25];
  const float* pred_b1   = (const float*)d_in[26];
  const float* pred_W2   = (const float*)d_in[27];
  const float* pred_b2   = (const float*)d_in[28];

  const int N = in_sizes[0];
  const int E = in_sizes[1] / 2;
  const int B = out_size;               // num_tasks == 1
  const int H = in_sizes[6];            // 256
  const int AD = in_sizes[5] / H;       // 128
  const int L = in_sizes[7];            // 5
  const int H2 = in_sizes[26];          // 128

  float* ws = (float*)d_ws;
  const size_t NH = (size_t)N * H;
  const size_t BH = (size_t)B * H;
  float* x      = ws;
  float* h      = x + NH;
  float* agg    = h + NH;
  float* s      = agg + NH;
  float* meanb  = s + BH;
  float* vn     = meanb + BH;
  float* yv     = vn + BH;              // also z / hidden scratch
  float* counts = yv + BH;
  float* colsum = counts + B;
  float* colsq  = colsum + H;
  float* wsum   = colsq + H;
  float* wsq    = wsum + H;

  const int gM = ceil_div_l(N, 64);     // workgroups in M for node GEMMs
  const int gB = ceil_div_l(B, 64);     // workgroups in M for graph GEMMs
  const int tN = H / 128;               // N-direction workgroups (H=256 -> 2)

  // per-graph node counts (once)
  hipMemsetAsync(counts, 0, (size_t)B * sizeof(float), stream);
  counts_kernel<<<ceil_div_l(N, 256), 256, 0, stream>>>(batch, counts, N);

  // x = emb_table[atom_x] @ proj_W + proj_b
  gather_emb_kernel<<<ceil_div_l((long)N * AD, 256), 256, 0, stream>>>(
      atom_x, emb_table, h, (long)N * AD, AD);
  gemm_wmma_f32<false, false><<<gM * tN, 256, 0, stream>>>(
      h, proj_W, proj_b, x, N, AD, H, nullptr, nullptr);

  // vn = broadcast(vn_init)
  vninit_kernel<<<ceil_div_l((long)BH, 256), 256, 0, stream>>>(vn, vn_init,
                                                               (long)BH, H);

  for (int i = 0; i < L; ++i) {
    // GIN aggregation
    hipMemsetAsync(agg, 0, NH * sizeof(float), stream);
    edge_agg_kernel<<<ceil_div_l((long)E * 32, 256), 256, 0, stream>>>(
        edge_index, edge_index + E, x, agg, E, H);
    hpre_kernel<<<ceil_div_l((long)NH, 256), 256, 0, stream>>>(x, agg, gin_eps,
                                                               i, (long)NH);
    // Linear1 + BN + ReLU
    hipMemsetAsync(colsum, 0, 2 * (size_t)H * sizeof(float), stream);
    gemm_wmma_f32<true, false><<<gM * tN, 256, 0, stream>>>(
        agg, gin_W1 + (size_t)i * H * H, gin_b1 + (size_t)i * H, h, N, H, H,
        colsum, colsq);
    bn_relu_kernel<<<ceil_div_l((long)NH, 256), 256, 0, stream>>>(
        h, h, colsum, colsq, gin_g1 + (size_t)i * H, gin_bt1 + (size_t)i * H,
        (long)NH, H, 1.f / N);
    // Linear2 + BN + ReLU
    hipMemsetAsync(colsum, 0, 2 * (size_t)H * sizeof(float), stream);
    gemm_wmma_f32<true, false><<<gM * tN, 256, 0, stream>>>(
        h, gin_W2 + (size_t)i * H * H, gin_b2 + (size_t)i * H, x, N, H, H,
        colsum, colsq);
    bn_relu_kernel<<<ceil_div_l((long)NH, 256), 256, 0, stream>>>(
        x, x, colsum, colsq, gin_g2 + (size_t)i * H, gin_bt2 + (size_t)i * H,
        (long)NH, H, 1.f / N);

    if (i < L - 1) {
      // node -> virtual node
      hipMemsetAsync(s, 0, BH * sizeof(float), stream);
      seg_sum_kernel<<<ceil_div_l((long)N * 32, 256), 256, 0, stream>>>(
          batch, x, s, N, H);
      mean_kernel<<<ceil_div_l((long)BH, 256), 256, 0, stream>>>(
          s, counts, meanb, (long)BH, H);
      hipMemsetAsync(colsum, 0, 2 * (size_t)H * sizeof(float), stream);
      gemm_wmma_f32<true, false><<<gB * tN, 256, 0, stream>>>(
          meanb, n2v_W + (size_t)i * H * H, n2v_b + (size_t)i * H, yv, B, H, H,
          colsum, colsq);
      vn_acc_kernel<<<ceil_div_l((long)BH, 256), 256, 0, stream>>>(
          vn, yv, colsum, colsq, n2v_g + (size_t)i * H, n2v_bt + (size_t)i * H,
          (long)BH, H, 1.f / B);
      // virtual node -> node (GEMM at B granularity; BN stats count-weighted)
      gemm_wmma_f32<false, false><<<gB * tN, 256, 0, stream>>>(
          vn, v2n_W + (size_t)i * H * H, v2n_b + (size_t)i * H, yv, B, H, H,
          nullptr, nullptr);
      hipMemsetAsync(wsum, 0, 2 * (size_t)H * sizeof(float), stream);
      wstats_kernel<<<16, H, 0, stream>>>(yv, counts, wsum, wsq, B, H);
      v2n_apply_kernel<<<ceil_div_l((long)N * 32, 256), 256, 0, stream>>>(
          x, yv, batch, wsum, wsq, v2n_g + (size_t)i * H,
          v2n_bt + (size_t)i * H, N, H, 1.f / N);
    }
  }

  // global sum pooling + prediction head
  hipMemsetAsync(s, 0, BH * sizeof(float), stream);
  seg_sum_kernel<<<ceil_div_l((long)N * 32, 256), 256, 0, stream>>>(batch, x, s,
                                                                    N, H);
  gemm_wmma_f32<false, true><<<gB * (H2 / 128), 256, 0, stream>>>(
      s, pred_W1, pred_b1, yv, B, H, H2, nullptr, nullptr);
  pred2_kernel<<<ceil_div_l(B, 256), 256, 0, stream>>>(yv, pred_W2, pred_b2,
                                                       (float*)d_out, B, H2);
}